// SimpleGraphClassifier_9775345566347
// MI455X (gfx1250) — compile-verified
//
#include <hip/hip_runtime.h>
#include <hip/hip_bf16.h>

// Problem constants (match reference)
#define NN 16384      // nodes
#define EE 524288     // edges
#define DD 128        // node feature dim
#define HH 256        // hidden dim
#define CC 10         // classes
#define GG 64         // graphs

typedef __attribute__((ext_vector_type(2))) float v2f;
typedef __attribute__((ext_vector_type(8))) float v8f;

// ---------------------------------------------------------------------------
// 1) Count out-edges per row (adjacency row = edge_index[0])
// ---------------------------------------------------------------------------
__global__ __launch_bounds__(256) void count_edges_kernel(
    const int* __restrict__ edge_index, int* __restrict__ row_counts) {
  int e = blockIdx.x * blockDim.x + threadIdx.x;
  if (e < EE) {
    int r = edge_index[e];  // edge_index[0][e]
    atomicAdd(&row_counts[r], 1);
  }
}

// ---------------------------------------------------------------------------
// 2) Exclusive scan of (1 + count) -> row_start[NN+1].  Single block.
// ---------------------------------------------------------------------------
__global__ __launch_bounds__(256) void scan_kernel(
    const int* __restrict__ row_counts, int* __restrict__ row_start) {
  __shared__ int sums[256];
  const int PER = NN / 256;  // 64
  int t = threadIdx.x;
  int local[PER];
  int s = 0;
  for (int i = 0; i < PER; ++i) {
    local[i] = s;
    s += 1 + row_counts[t * PER + i];
  }
  sums[t] = s;
  __syncthreads();
  if (t == 0) {
    int acc = 0;
    for (int i = 0; i < 256; ++i) { int v = sums[i]; sums[i] = acc; acc += v; }
    row_start[NN] = acc;
  }
  __syncthreads();
  int base = sums[t];
  for (int i = 0; i < PER; ++i) row_start[t * PER + i] = base + local[i];
}

// ---------------------------------------------------------------------------
// 3) Seed each row with its self-loop column; init append cursor
// ---------------------------------------------------------------------------
__global__ __launch_bounds__(256) void fill_self_kernel(
    const int* __restrict__ row_start, int* __restrict__ csr_col,
    int* __restrict__ cursor) {
  int i = blockIdx.x * blockDim.x + threadIdx.x;
  if (i < NN) {
    int s = row_start[i];
    csr_col[s] = i;
    cursor[i] = s + 1;
  }
}

// ---------------------------------------------------------------------------
// 4) Append edge destination columns
// ---------------------------------------------------------------------------
__global__ __launch_bounds__(256) void fill_edges_kernel(
    const int* __restrict__ edge_index, int* __restrict__ cursor,
    int* __restrict__ csr_col) {
  int e = blockIdx.x * blockDim.x + threadIdx.x;
  if (e < EE) {
    int r = edge_index[e];
    int c = edge_index[EE + e];  // edge_index[1][e]
    int pos = atomicAdd(&cursor[r], 1);
    csr_col[pos] = c;
  }
}

// ---------------------------------------------------------------------------
// 5) Per-row insertion sort + dedupe (reference collapses duplicates to 1).
//    deg = unique count (self-loop guarantees >= 1); dis = deg^-1/2.
// ---------------------------------------------------------------------------
__global__ __launch_bounds__(256) void sort_dedupe_kernel(
    const int* __restrict__ row_start, int* __restrict__ csr_col,
    int* __restrict__ row_len, float* __restrict__ dis) {
  int i = blockIdx.x * blockDim.x + threadIdx.x;
  if (i >= NN) return;
  int start = row_start[i];
  int end = row_start[i + 1];
  for (int a = start + 1; a < end; ++a) {
    int v = csr_col[a];
    int b = a - 1;
    while (b >= start && csr_col[b] > v) { csr_col[b + 1] = csr_col[b]; --b; }
    csr_col[b + 1] = v;
  }
  int u = 1;
  for (int a = start + 1; a < end; ++a) {
    int v = csr_col[a];
    if (v != csr_col[start + u - 1]) { csr_col[start + u] = v; ++u; }
  }
  row_len[i] = u;
  dis[i] = 1.0f / sqrtf((float)u);
}

// ---------------------------------------------------------------------------
// 6) Sparse aggregation: y[i][:] = dis[i] * sum_j dis[j] * x[j][:]
//    One block per row, one thread per feature dim; neighbor list staged
//    through LDS in chunks of 128. x fits in L2 (8-16MB << 192MB).
// ---------------------------------------------------------------------------
template <int DIM>
__global__ __launch_bounds__(DIM) void spmm_kernel(
    const float* __restrict__ x, const int* __restrict__ csr_col,
    const int* __restrict__ row_start, const int* __restrict__ row_len,
    const float* __restrict__ dis, float* __restrict__ y) {
  __shared__ int s_col[128];
  __shared__ float s_dis[128];
  int row = blockIdx.x;
  int d = threadIdx.x;
  int base = row_start[row];
  int len = row_len[row];
  float acc = 0.0f;
  for (int t0 = 0; t0 < len; t0 += 128) {
    int nload = len - t0; if (nload > 128) nload = 128;
    if ((int)threadIdx.x < nload) {
      int c = csr_col[base + t0 + threadIdx.x];
      s_col[threadIdx.x] = c;
      s_dis[threadIdx.x] = dis[c];
    }
    __syncthreads();
    for (int t = 0; t < nload; ++t) {
      acc += s_dis[t] * x[(size_t)s_col[t] * DIM + d];
    }
    __syncthreads();
  }
  y[(size_t)row * DIM + d] = dis[row] * acc;
}

// ---------------------------------------------------------------------------
// 7) Dense GEMM + bias + ReLU using V_WMMA_F32_16X16X4_F32 (fp32 matrix op).
//    O[M x 256] = relu(A[M x K] @ B[K x 256] + bias).
//    Per ISA layouts:
//      A 16x4 : lane m = lane&15; K-half = (lane>>4)*2 + vgpr{0,1}
//      B 4x16 : lane n = lane&15; K-half = (lane>>4)*2 + vgpr{0,1}
//      C 16x16: vgpr v -> row v (lanes 0-15) / v+8 (lanes 16-31), col = lane&15
//    4 waves per block, each wave owns one 16x16 output tile.
// ---------------------------------------------------------------------------
template <int K>
__global__ __launch_bounds__(128) void gemm_bias_relu_kernel(
    const float* __restrict__ A, const float* __restrict__ B,
    const float* __restrict__ bias, float* __restrict__ O) {
  const int NT = HH / 16;  // 16 tiles along N
  int wave = threadIdx.x >> 5;
  int lane = threadIdx.x & 31;
  int tile = blockIdx.x * 4 + wave;
  int tm = tile / NT;
  int tn = tile % NT;
  int row0 = tm * 16;
  int col0 = tn * 16;
  int mn = lane & 15;           // A row within tile / B col within tile
  int kh = (lane >> 4) * 2;     // K sub-offset for this lane half

  const float* arow = A + (size_t)(row0 + mn) * K + kh;
  const float* bcol = B + (size_t)kh * HH + (col0 + mn);

  v8f c = {};
  for (int k0 = 0; k0 < K; k0 += 4) {
    v2f a;
    a.x = arow[k0];
    a.y = arow[k0 + 1];
    v2f b;
    b.x = bcol[(size_t)k0 * HH];
    b.y = bcol[(size_t)(k0 + 1) * HH];
    // (neg_a, A, neg_b, B, c_mod, C, reuse_a, reuse_b)
    c = __builtin_amdgcn_wmma_f32_16x16x4_f32(false, a, false, b, (short)0, c,
                                              false, false);
  }

  int col = col0 + (lane & 15);
  int rowAdd = (lane >> 4) * 8;
  float bv = bias[col];
#pragma unroll
  for (int v = 0; v < 8; ++v) {
    float val = c[v] + bv;
    val = fmaxf(val, 0.0f);
    O[(size_t)(row0 + v + rowAdd) * HH + col] = val;
  }
}

// ---------------------------------------------------------------------------
// 8) Mean pooling: per-graph counts + feature sums via atomics
// ---------------------------------------------------------------------------
__global__ __launch_bounds__(256) void count_batch_kernel(
    const int* __restrict__ batch, int* __restrict__ counts) {
  int i = blockIdx.x * blockDim.x + threadIdx.x;
  if (i < NN) atomicAdd(&counts[batch[i]], 1);
}

__global__ __launch_bounds__(256) void pool_sum_kernel(
    const float* __restrict__ h, const int* __restrict__ batch,
    float* __restrict__ pooled) {
  int i = blockIdx.x;   // node
  int d = threadIdx.x;  // feature
  int g = batch[i];
  atomicAdd(&pooled[(size_t)g * HH + d], h[(size_t)i * HH + d]);
}

// ---------------------------------------------------------------------------
// 9) Classifier: out[g][c] = (pooled[g]/count[g]) @ Wc + bc
// ---------------------------------------------------------------------------
__global__ __launch_bounds__(256) void classify_kernel(
    const float* __restrict__ pooled, const int* __restrict__ counts,
    const float* __restrict__ Wc, const float* __restrict__ bc,
    float* __restrict__ out) {
  int idx = blockIdx.x * blockDim.x + threadIdx.x;
  if (idx >= GG * CC) return;
  int g = idx / CC;
  int c = idx % CC;
  float inv = 1.0f / fmaxf((float)counts[g], 1.0f);
  float acc = bc[c];
  for (int hh = 0; hh < HH; ++hh) {
    acc += (pooled[(size_t)g * HH + hh] * inv) * Wc[hh * CC + c];
  }
  out[idx] = acc;
}

// ---------------------------------------------------------------------------
// Launch
// ---------------------------------------------------------------------------
extern "C" void kernel_launch(void* const* d_in, const int* in_sizes, int n_in,
                              void* d_out, int out_size, void* d_ws,
                              size_t ws_size, hipStream_t stream) {
  const float* x = (const float*)d_in[0];          // [N, D]
  const int* edge_index = (const int*)d_in[1];     // [2, E]
  const int* batch = (const int*)d_in[2];          // [N]
  const float* W1 = (const float*)d_in[3];         // [D, H]
  const float* b1 = (const float*)d_in[4];         // [H]
  const float* W2 = (const float*)d_in[5];         // [H, H]
  const float* b2 = (const float*)d_in[6];         // [H]
  const float* Wc = (const float*)d_in[7];         // [H, C]
  const float* bc = (const float*)d_in[8];         // [C]
  float* out = (float*)d_out;                      // [G, C]

  // Workspace layout (256B aligned slices)
  char* base = (char*)d_ws;
  size_t off = 0;
  auto alloc = [&](size_t bytes) -> void* {
    void* p = base + off;
    off = (off + bytes + 255) & ~(size_t)255;
    return p;
  };
  int* csr_col = (int*)alloc((size_t)(NN + EE) * 4);
  int* row_start = (int*)alloc((size_t)(NN + 1) * 4);
  int* row_len = (int*)alloc((size_t)NN * 4);
  int* cursor = (int*)alloc((size_t)NN * 4);
  int* row_counts = (int*)alloc((size_t)NN * 4);
  float* dis = (float*)alloc((size_t)NN * 4);
  int* counts = (int*)alloc((size_t)GG * 4);
  float* pooled = (float*)alloc((size_t)GG * HH * 4);
  float* bufA = (float*)alloc((size_t)NN * HH * 4);  // Y1 (N x D) then Y2 (N x H)
  float* bufB = (float*)alloc((size_t)NN * HH * 4);  // H1 then H2
  (void)ws_size;

  // Zero accumulators (async memset is graph-capturable)
  hipMemsetAsync(row_counts, 0, (size_t)NN * 4, stream);
  hipMemsetAsync(counts, 0, (size_t)GG * 4, stream);
  hipMemsetAsync(pooled, 0, (size_t)GG * HH * 4, stream);

  // Build deduped CSR of (A + I)
  count_edges_kernel<<<(EE + 255) / 256, 256, 0, stream>>>(edge_index, row_counts);
  scan_kernel<<<1, 256, 0, stream>>>(row_counts, row_start);
  fill_self_kernel<<<(NN + 255) / 256, 256, 0, stream>>>(row_start, csr_col, cursor);
  fill_edges_kernel<<<(EE + 255) / 256, 256, 0, stream>>>(edge_index, cursor, csr_col);
  sort_dedupe_kernel<<<(NN + 255) / 256, 256, 0, stream>>>(row_start, csr_col,
                                                           row_len, dis);

  // Layer 1: Y1 = Anorm @ x ; H1 = relu(Y1 @ W1 + b1)
  spmm_kernel<DD><<<NN, DD, 0, stream>>>(x, csr_col, row_start, row_len, dis, bufA);
  gemm_bias_relu_kernel<DD>
      <<<(NN / 16) * (HH / 16) / 4, 128, 0, stream>>>(bufA, W1, b1, bufB);

  // Layer 2: Y2 = Anorm @ H1 ; H2 = relu(Y2 @ W2 + b2)
  spmm_kernel<HH><<<NN, HH, 0, stream>>>(bufB, csr_col, row_start, row_len, dis, bufA);
  gemm_bias_relu_kernel<HH>
      <<<(NN / 16) * (HH / 16) / 4, 128, 0, stream>>>(bufA, W2, b2, bufB);

  // Mean pool + classify
  count_batch_kernel<<<(NN + 255) / 256, 256, 0, stream>>>(batch, counts);
  pool_sum_kernel<<<NN, HH, 0, stream>>>(bufB, batch, pooled);
  classify_kernel<<<(GG * CC + 255) / 256, 256, 0, stream>>>(pooled, counts, Wc,
                                                             bc, out);
}